// GLR_37881611550877
// MI455X (gfx1250) — compile-verified
//
#include <hip/hip_runtime.h>
#include <hip/hip_bf16.h>
#include <math.h>

// Problem constants (match reference)
#define Bc 2
#define Gc 4
#define Fc 9
#define Cc 3
#define Hc 512
#define Wc 512
#define Nc (Hc * Wc)
#define BGc (Bc * Gc)
#define FP 12              // fM record padded to 12 floats = 3 x float4 (48B)
#define TILES_PER_BG (Nc / 16)

typedef float v2f __attribute__((ext_vector_type(2)));
typedef float v8f __attribute__((ext_vector_type(8)));

// ---------------- Kernel R: build CSR row pointers from sorted edge list ----
// edges are sorted by nodeI and every node has a self-edge, so every i appears.
__global__ void GLR_build_rowptr(const int* __restrict__ edges, int E,
                                 int* __restrict__ row_ptr) {
    int e = blockIdx.x * blockDim.x + threadIdx.x;
    if (e >= E) return;
    int i = edges[2 * e];
    if (e == 0 || edges[2 * (e - 1)] != i) row_ptr[i] = e;
    if (e == 0) row_ptr[Nc] = E;
}

// ---------------- Kernel A: normalize features + apply multiM via f32 WMMA --
// Per wave: 16 nodes. D(16x16) = A(16x4) x B(4x16) + C over 3 K-chunks.
// A = multiM^T (rows v, cols c), B = normalized features (rows c, cols node).
__global__ __launch_bounds__(128)
void GLR_feat_wmma(const float* __restrict__ img, const float* __restrict__ multiM,
                   const int* __restrict__ order, float4* __restrict__ fM4) {
    __shared__ float lds[4][16][FP];

    const int lane = threadIdx.x & 31;
    const int wave = threadIdx.x >> 5;
    const int tile = blockIdx.x * 4 + wave;
    const int bg   = tile / TILES_PER_BG;
    const int base = (tile % TILES_PER_BG) * 16;
    const int g    = bg % Gc;

    const int v    = lane & 15;   // A row / B column / D half-row select
    const int half = lane >> 4;

    // A-matrix regs (constant across node tiles): A[v][c] = multiM[g][c][v]
    // ISA layout: vgpr p, lanes0-15 -> K=p, lanes16-31 -> K=p+2
    v2f A[3];
    #pragma unroll
    for (int k = 0; k < 3; ++k) {
        #pragma unroll
        for (int p = 0; p < 2; ++p) {
            int c = k * 4 + p + 2 * half;
            float a = 0.0f;
            if (v < Fc && c < Fc) a = multiM[(g * Fc + c) * Fc + v];
            A[k][p] = a;
        }
    }

    // Stage: lanes 0-15 gather one node's 9 channels, normalize, write LDS
    if (lane < 16) {
        int n = base + lane;
        int o = order[n];
        const float* ip = img + (size_t)bg * Fc * Nc + o;
        float f[Fc];
        float s = 0.0f;
        #pragma unroll
        for (int c = 0; c < Fc; ++c) {
            f[c] = ip[(size_t)c * Nc];
            s += f[c] * f[c];
        }
        float inv = 1.0f / fmaxf(sqrtf(s), 1e-12f);
        #pragma unroll
        for (int c = 0; c < Fc; ++c) lds[wave][lane][c] = f[c] * inv;
        lds[wave][lane][9]  = 0.0f;
        lds[wave][lane][10] = 0.0f;
        lds[wave][lane][11] = 0.0f;
    }
    __syncthreads();

    // 3 chained f32 WMMAs over K-chunks {0-3},{4-7},{8-11}
    v8f acc = {};
    #pragma unroll
    for (int k = 0; k < 3; ++k) {
        v2f Bm;
        Bm[0] = lds[wave][v][k * 4 + 0 + 2 * half];
        Bm[1] = lds[wave][v][k * 4 + 1 + 2 * half];
        acc = __builtin_amdgcn_wmma_f32_16x16x4_f32(
            /*neg_a=*/false, A[k], /*neg_b=*/false, Bm,
            /*c_mod=*/(short)0, acc, /*reuse_a=*/false, /*reuse_b=*/false);
    }

    // D layout: vgpr r, lanes0-15 -> row v=r, lanes16-31 -> row v=r+8.
    // Rows 9..11 are exact zeros (zero A rows) and serve as record padding.
    // Uniform first store keeps both stores as aligned b128:
    //   half==0: out4[0]=acc[0..3], out4[1]=acc[4..7]   (rows 0..7)
    //   half==1: out4[2]=acc[0..3]                      (rows 8..11, 9..11 = 0)
    float4* out4 = fM4 + ((size_t)bg * Nc + base + v) * 3 + 2 * half;
    out4[0] = make_float4(acc[0], acc[1], acc[2], acc[3]);
    if (half == 0)
        out4[1] = make_float4(acc[4], acc[5], acc[6], acc[7]);
}

__device__ __forceinline__ float GLR_dot9(const float4 x0, const float4 x1, float x8,
                                          const float4* __restrict__ rec) {
    float4 y0 = rec[0];
    float4 y1 = rec[1];
    float  y8 = ((const float*)rec)[8];
    return x0.x * y0.x + x0.y * y0.y + x0.z * y0.z + x0.w * y0.w
         + x1.x * y1.x + x1.y * y1.y + x1.z * y1.z + x1.w * y1.w
         + x8 * y8;
}

// ---------------- Kernel B: per-node degree -> dinv (no atomics via CSR) ----
__global__ void GLR_edge_deg(const float4* __restrict__ fM4, const int* __restrict__ edges,
                             const int* __restrict__ row_ptr, float* __restrict__ dinv) {
    int t = blockIdx.x * blockDim.x + threadIdx.x;
    if (t >= BGc * Nc) return;
    int bg = t / Nc;
    int i  = t % Nc;
    const float4* fbase = fM4 + (size_t)bg * Nc * 3;
    const float4* pi = fbase + (size_t)i * 3;
    float4 x0 = pi[0];
    float4 x1 = pi[1];
    float  x8 = ((const float*)pi)[8];
    int e0 = row_ptr[i], e1 = row_ptr[i + 1];
    float deg = 0.0f;
    for (int e = e0; e < e1; ++e) {
        int j = edges[2 * e + 1];
        float sim = GLR_dot9(x0, x1, x8, fbase + (size_t)j * 3);
        sim = fminf(fmaxf(sim, -10.0f), 10.0f);
        deg += __expf(sim);
    }
    dinv[t] = rsqrtf(deg);
}

// ---------------- Kernel C: aggregate signals + write output ----------------
__global__ void GLR_aggregate(const float4* __restrict__ fM4, const float* __restrict__ sig,
                              const int* __restrict__ edges, const int* __restrict__ row_ptr,
                              const float* __restrict__ dinv, const int* __restrict__ order,
                              float* __restrict__ out) {
    int t = blockIdx.x * blockDim.x + threadIdx.x;
    if (t >= BGc * Nc) return;
    int bg = t / Nc;
    int i  = t % Nc;
    const float4* fbase = fM4 + (size_t)bg * Nc * 3;
    const float4* pi = fbase + (size_t)i * 3;
    float4 x0 = pi[0];
    float4 x1 = pi[1];
    float  x8 = ((const float*)pi)[8];
    const float* dv = dinv + (size_t)bg * Nc;
    const float di = dv[i];
    const float* sbase = sig + (size_t)bg * Cc * Nc;

    int e0 = row_ptr[i], e1 = row_ptr[i + 1];
    float a0 = 0.0f, a1 = 0.0f, a2 = 0.0f;
    for (int e = e0; e < e1; ++e) {
        int j = edges[2 * e + 1];
        float sim = GLR_dot9(x0, x1, x8, fbase + (size_t)j * 3);
        sim = fminf(fmaxf(sim, -10.0f), 10.0f);
        float w = di * __expf(sim) * dv[j];
        int oj = order[j];
        a0 += w * sbase[0 * Nc + oj];
        a1 += w * sbase[1 * Nc + oj];
        a2 += w * sbase[2 * Nc + oj];
    }
    int oi = order[i];  // out2d[..., order[i]] = signals2d[..., order[i]] - agg[i]
    float* obase = out + (size_t)bg * Cc * Nc;
    obase[0 * Nc + oi] = sbase[0 * Nc + oi] - a0;
    obase[1 * Nc + oi] = sbase[1 * Nc + oi] - a1;
    obase[2 * Nc + oi] = sbase[2 * Nc + oi] - a2;
}

extern "C" void kernel_launch(void* const* d_in, const int* in_sizes, int n_in,
                              void* d_out, int out_size, void* d_ws, size_t ws_size,
                              hipStream_t stream) {
    (void)n_in; (void)out_size; (void)ws_size;
    const float* img   = (const float*)d_in[0];   // (B,G,F,H,W) f32
    const float* sig   = (const float*)d_in[1];   // (B,G,C,H,W) f32
    const float* mM    = (const float*)d_in[2];   // (G,F,F) f32
    const int*   edges = (const int*)d_in[3];     // (E,2) i32, sorted by nodeI
    const int*   order = (const int*)d_in[4];     // (N,) i32
    // d_in[5] order_inverse unused (inverted analytically)
    const int E = in_sizes[3] / 2;
    float* out = (float*)d_out;

    // Workspace partition: fM (48B/node/bg) | dinv | row_ptr  (~110 MB total)
    char* w = (char*)d_ws;
    float4* fM4 = (float4*)w;
    size_t fM_b = (size_t)BGc * Nc * FP * sizeof(float);
    float* dinv = (float*)(w + fM_b);
    size_t dv_b = (size_t)BGc * Nc * sizeof(float);
    int* row_ptr = (int*)(w + fM_b + dv_b);

    GLR_build_rowptr<<<(E + 255) / 256, 256, 0, stream>>>(edges, E, row_ptr);

    // 131072 wave-tiles of 16 nodes, 4 waves (128 threads) per block
    GLR_feat_wmma<<<(BGc * TILES_PER_BG) / 4, 128, 0, stream>>>(img, mM, order, fM4);

    const int nt = BGc * Nc;
    GLR_edge_deg<<<(nt + 255) / 256, 256, 0, stream>>>(fM4, edges, row_ptr, dinv);
    GLR_aggregate<<<(nt + 255) / 256, 256, 0, stream>>>(fM4, sig, edges, row_ptr,
                                                        dinv, order, out);
}